// PseudoOneHotEncoding_51797305589978
// MI455X (gfx1250) — compile-verified
//
#include <hip/hip_runtime.h>

// PseudoOneHotEncoding: out[e, 0..20] = table[seq[e], 0..20]
// Pure gather, store-bandwidth bound (88 MB out @ 23.3 TB/s ~ 4.1 us floor).
// Strategy:
//   * table (2268 B) DMA'd into LDS via GLOBAL_LOAD_ASYNC_TO_LDS_B32
//     (gfx1250 async path, ASYNCcnt), no VGPR staging
//   * 1 thread = 1 float4 of output, branchless dual-code select
//   * coalesced global_store_b128 with th:TH_NT (output has no consumer;
//     keep the 192MB L2 for data with reuse)

#define NUM_CLASSES 27
#define ALPHABET    21
#define TBL_ELEMS   (NUM_CLASSES * ALPHABET)   // 567 floats = 2268 B

typedef float v4f __attribute__((ext_vector_type(4)));

__global__ __launch_bounds__(256)
void onehot_gather_f4(const int* __restrict__ seq,
                      const float* __restrict__ table,
                      v4f* __restrict__ out,
                      unsigned n_float4)
{
    __shared__ float lt[TBL_ELEMS];

    // Async memory->LDS fill: each active lane issues one
    // global_load_async_to_lds_b32 (GVS addressing: LDS-dst VGPR, 32-bit
    // voffset, 64-bit saddr). Low 32 bits of a generic LDS pointer are the
    // wave-logical LDS byte offset (aperture tag lives in the high bits).
    for (unsigned i = threadIdx.x; i < TBL_ELEMS; i += blockDim.x) {
        unsigned lds_off = (unsigned)(uintptr_t)&lt[i];
        unsigned goff    = i * 4u;
        asm volatile("global_load_async_to_lds_b32 %0, %1, %2"
                     :
                     : "v"(lds_off), "v"(goff), "s"(table)
                     : "memory");
    }
    asm volatile("s_wait_asynccnt 0" ::: "memory");
    __syncthreads();

    unsigned i4 = blockIdx.x * blockDim.x + threadIdx.x;
    if (i4 >= n_float4) return;

    unsigned flat = i4 * 4u;                  // flat float index into [E, 21]
    unsigned e0   = flat / 21u;               // magic-multiply, no hw div
    unsigned k0   = flat - e0 * 21u;
    // A float4 window spans at most 2 elements; it wraps iff k0+3 >= 21.
    unsigned e1   = e0 + (k0 >= 18u ? 1u : 0u);

    unsigned b0 = (unsigned)seq[e0] * (unsigned)ALPHABET;  // L0 hits: heavy
    unsigned b1 = (unsigned)seq[e1] * (unsigned)ALPHABET;  // intra-wave reuse

    v4f o;
#pragma unroll
    for (int j = 0; j < 4; ++j) {
        unsigned k   = k0 + (unsigned)j;
        unsigned idx = (k >= (unsigned)ALPHABET) ? (b1 + k - (unsigned)ALPHABET)
                                                 : (b0 + k);
        o[j] = lt[idx];                        // branchless: cmp + cndmask + ds_load
    }

    __builtin_nontemporal_store(o, out + i4);  // global_store_b128 th:TH_NT
}

// Scalar tail for out_size % 4 != 0 (not hit for 512x2048x21, kept for safety).
__global__ void onehot_gather_tail(const int* __restrict__ seq,
                                   const float* __restrict__ table,
                                   float* __restrict__ out,
                                   unsigned start_flat, unsigned total_flat)
{
    unsigned flat = start_flat + blockIdx.x * blockDim.x + threadIdx.x;
    if (flat >= total_flat) return;
    unsigned e = flat / 21u;
    unsigned k = flat - e * 21u;
    unsigned code = (unsigned)seq[e];
    __builtin_nontemporal_store(table[code * (unsigned)ALPHABET + k], out + flat);
}

extern "C" void kernel_launch(void* const* d_in, const int* in_sizes, int n_in,
                              void* d_out, int out_size, void* d_ws, size_t ws_size,
                              hipStream_t stream)
{
    const int*   seq   = (const int*)d_in[0];     // [512, 2048] int codes
    const float* table = (const float*)d_in[1];   // [27, 21] f32
    float*       out   = (float*)d_out;           // [512, 2048, 21] f32

    const unsigned total    = (unsigned)out_size;
    const unsigned n_float4 = total / 4u;
    const unsigned rem      = total - n_float4 * 4u;

    if (n_float4 > 0) {
        dim3 block(256);
        dim3 grid((n_float4 + 255u) / 256u);
        onehot_gather_f4<<<grid, block, 0, stream>>>(seq, table,
                                                     (v4f*)out, n_float4);
    }
    if (rem > 0) {
        onehot_gather_tail<<<1, 32, 0, stream>>>(seq, table, out,
                                                 n_float4 * 4u, total);
    }
}